// MPNNBlock_5282809774572
// MI455X (gfx1250) — compile-verified
//
#include <hip/hip_runtime.h>

#define N_NODES 100000
#define N_EDGES 800000
#define HDIM 128
#define EDIM 32
#define FAN  288   // 2*H + E
#define NLAYER 3

typedef float v8f __attribute__((ext_vector_type(8)));
typedef float v2f __attribute__((ext_vector_type(2)));

__device__ __forceinline__ v2f ld2(const float* p) { return *(const v2f*)p; }

// D = A(16x4 f32) * B(4x16 f32) + C(16x16 f32), full fp32 tensor op on CDNA5
__device__ __forceinline__ v8f wmma4(v2f a, v2f b, v8f c) {
    return __builtin_amdgcn_wmma_f32_16x16x4_f32(false, a, false, b, (short)0, c,
                                                 false, false);
}

// ---------------- degree ----------------
__global__ void zero_deg_kernel(float* __restrict__ deg) {
    int i = blockIdx.x * blockDim.x + threadIdx.x;
    if (i < N_NODES) deg[i] = 0.f;
}

__global__ void count_deg_kernel(const long long* __restrict__ dstI,
                                 float* __restrict__ deg) {
    int e = blockIdx.x * blockDim.x + threadIdx.x;
    if (e < N_EDGES) unsafeAtomicAdd(&deg[dstI[e]], 1.f);
}

// ---------------- node-level GEMM: A = X*Wsrc^T ; accum = deg*(X*Wdst^T + b) ----------------
__global__ void __launch_bounds__(256)
node_gemm_kernel(const float* __restrict__ xin,
                 const float* __restrict__ W,      // [128 x 288] row-major (out, in)
                 const float* __restrict__ bias,   // [128]
                 const float* __restrict__ deg,    // [N]
                 float* __restrict__ Abuf,         // [N x 128]
                 float* __restrict__ accum)        // [N x 128]
{
    const int wave = (int)((blockIdx.x * blockDim.x + threadIdx.x) >> 5);
    if (wave >= N_NODES / 16) return;
    const int lane = threadIdx.x & 31;
    const int hi   = lane >> 4;
    const int l15  = lane & 15;
    const int nbase = wave * 16;

    v8f S[8], T[8];
    const v8f vzero = {0.f,0.f,0.f,0.f,0.f,0.f,0.f,0.f};
#pragma unroll
    for (int nt = 0; nt < 8; ++nt) { S[nt] = vzero; T[nt] = vzero; }

    const float* xrow = xin + (size_t)(nbase + l15) * HDIM;
    const float* wrow = W + (size_t)l15 * FAN;

#pragma unroll 4
    for (int k = 0; k < HDIM; k += 4) {
        const int ka = k + 2 * hi;
        const v2f a = ld2(xrow + ka);               // A operand: 16x4 chunk of X
#pragma unroll
        for (int nt = 0; nt < 8; ++nt) {
            const float* wr = wrow + (size_t)(nt * 16) * FAN;
            const v2f bs = ld2(wr + ka);            // Wsrc cols [0,128)
            const v2f bd = ld2(wr + 128 + ka);      // Wdst cols [128,256)
            S[nt] = wmma4(a, bs, S[nt]);
            T[nt] = wmma4(a, bd, T[nt]);
        }
    }

    float dg[8];
#pragma unroll
    for (int r = 0; r < 8; ++r) dg[r] = deg[nbase + r + 8 * hi];

#pragma unroll
    for (int nt = 0; nt < 8; ++nt) {
        const int chan = nt * 16 + l15;
        const float bc = bias[chan];
#pragma unroll
        for (int r = 0; r < 8; ++r) {
            const int row = nbase + r + 8 * hi;     // C/D layout: VGPR r -> M=r (lo) / r+8 (hi)
            Abuf[(size_t)row * HDIM + chan]  = S[nt][r];
            accum[(size_t)row * HDIM + chan] = dg[r] * (T[nt][r] + bc);
        }
    }
}

// ---------------- edge GEMM + fused gather/scatter ----------------
// acc = A[src] (gathered into C operand) + e_attr * We^T ; atomic-add into accum[dst]
__global__ void __launch_bounds__(256)
edge_gemm_scatter_kernel(const float* __restrict__ eattr,
                         const long long* __restrict__ srcI,
                         const long long* __restrict__ dstI,
                         const float* __restrict__ W,   // layer base; We = cols [256,288)
                         const float* __restrict__ Abuf,
                         float* __restrict__ accum)
{
    __shared__ __align__(16) float lwe[HDIM * 34];   // We staged; stride 34 -> conflict-free b64
    for (int idx = threadIdx.x; idx < HDIM * EDIM; idx += 256) {
        const int o = idx >> 5, k = idx & 31;
        lwe[o * 34 + k] = W[(size_t)o * FAN + 256 + k];
    }
    __syncthreads();

    const int wave = (int)((blockIdx.x * blockDim.x + threadIdx.x) >> 5);
    if (wave >= N_EDGES / 16) return;
    const int lane = threadIdx.x & 31;
    const int hi   = lane >> 4;
    const int l15  = lane & 15;
    const int ebase = wave * 16;

    long long srow[8];
#pragma unroll
    for (int r = 0; r < 8; ++r) srow[r] = srcI[ebase + r + 8 * hi];

    // init accumulators with gathered A[src] rows (fuses the gather into the MMA)
    v8f acc[8];
#pragma unroll
    for (int nt = 0; nt < 8; ++nt) {
#pragma unroll
        for (int r = 0; r < 8; ++r)
            acc[nt][r] = Abuf[(size_t)srow[r] * HDIM + nt * 16 + l15];
    }

    const float* arow = eattr + (size_t)(ebase + l15) * EDIM;
#pragma unroll
    for (int k = 0; k < EDIM; k += 4) {
        const int ka = k + 2 * hi;
        const v2f a = ld2(arow + ka);
#pragma unroll
        for (int nt = 0; nt < 8; ++nt) {
            const v2f b = *(const v2f*)&lwe[(nt * 16 + l15) * 34 + ka];
            acc[nt] = wmma4(a, b, acc[nt]);
        }
    }

    long long drow[8];
#pragma unroll
    for (int r = 0; r < 8; ++r) drow[r] = dstI[ebase + r + 8 * hi];

#pragma unroll
    for (int nt = 0; nt < 8; ++nt) {
#pragma unroll
        for (int r = 0; r < 8; ++r)
            unsafeAtomicAdd(&accum[(size_t)drow[r] * HDIM + nt * 16 + l15],
                            acc[nt][r]);
    }
}

// ---------------- activation ----------------
__global__ void activate_kernel(const float* __restrict__ in, float* __restrict__ out,
                                int twice) {
    const long long i = (long long)blockIdx.x * blockDim.x + threadIdx.x;
    if (i >= (long long)N_NODES * HDIM) return;
    float v = in[i];
    v = v > 0.f ? v : 0.01f * v;
    if (twice) v = v > 0.f ? v : 0.01f * v;   // last layer: in-loop + final leaky_relu
    out[i] = v;
}

extern "C" void kernel_launch(void* const* d_in, const int* in_sizes, int n_in,
                              void* d_out, int out_size, void* d_ws, size_t ws_size,
                              hipStream_t stream) {
    const float*     x0    = (const float*)d_in[0];
    const long long* eidx  = (const long long*)d_in[1];   // int64 [2, E]
    const float*     eattr = (const float*)d_in[2];
    const float*     Wm    = (const float*)d_in[3];       // [3,128,288]
    const float*     bm    = (const float*)d_in[4];       // [3,128]
    float* out = (float*)d_out;

    float* deg = (float*)d_ws;                     // N floats (N%16==0, alignment ok)
    float* B0  = deg + N_NODES;                    // N*H
    float* B1  = B0 + (size_t)N_NODES * HDIM;      // N*H
    float* B2  = B1 + (size_t)N_NODES * HDIM;      // N*H (accum, reused per layer)

    const long long* srcI = eidx;
    const long long* dstI = eidx + N_EDGES;

    zero_deg_kernel<<<(N_NODES + 255) / 256, 256, 0, stream>>>(deg);
    count_deg_kernel<<<(N_EDGES + 255) / 256, 256, 0, stream>>>(dstI, deg);

    const int nodeBlocks = (N_NODES / 16 + 7) / 8;   // 782  (8 waves/block)
    const int edgeBlocks = (N_EDGES / 16 + 7) / 8;   // 6250
    const int actBlocks  = (N_NODES * HDIM + 255) / 256;

    const float* xin = x0;
    float* accum = B2;
    for (int l = 0; l < NLAYER; ++l) {
        float* Abuf = (l & 1) ? B1 : B0;
        const float* W = Wm + (size_t)l * HDIM * FAN;
        const float* b = bm + (size_t)l * HDIM;

        node_gemm_kernel<<<nodeBlocks, 256, 0, stream>>>(xin, W, b, deg, Abuf, accum);
        edge_gemm_scatter_kernel<<<edgeBlocks, 256, 0, stream>>>(eattr, srcI, dstI, W,
                                                                 Abuf, accum);
        float* xout = (l == NLAYER - 1) ? out : Abuf;   // Abuf dead after edge kernel
        activate_kernel<<<actBlocks, 256, 0, stream>>>(accum, xout,
                                                       (l == NLAYER - 1) ? 1 : 0);
        xin = xout;
    }
}